// GATLayer_45698452029991
// MI455X (gfx1250) — compile-verified
//
#include <hip/hip_runtime.h>
#include <hip/hip_bf16.h>
#include <math.h>

// GAT layer, MI455X (gfx1250, wave32).
// N=100000, E=1600000, IN=128, OUT=64, EDGE=16, HEADS=4, HEAD_DIM=16.

#define IN_DIM   128
#define OUT_DIM  64
#define EDGE_DIM 16
#define HEADS    4
#define HEAD_DIM 16

typedef float v2f __attribute__((ext_vector_type(2)));
typedef float v8f __attribute__((ext_vector_type(8)));

// ---- monotone float <-> uint encoding for atomic max over floats ----
__device__ __forceinline__ unsigned enc_f(float f) {
    unsigned u = __float_as_uint(f);
    return (u & 0x80000000u) ? ~u : (u | 0x80000000u);
}
__device__ __forceinline__ float dec_f(unsigned u) {
    return (u & 0x80000000u) ? __uint_as_float(u ^ 0x80000000u)
                             : __uint_as_float(~u);
}

// ---------------------------------------------------------------------------
// 0) init: zero out-accumulator, denominators, global max cell
// ---------------------------------------------------------------------------
__global__ void gat_init(float* __restrict__ out, float* __restrict__ den,
                         unsigned* __restrict__ gmax, int nOut, int nDen) {
    int t = blockIdx.x * blockDim.x + threadIdx.x;
    if (t < nOut) out[t] = 0.0f;
    if (t < nDen) den[t] = 0.0f;
    if (t == 0)   *gmax = 0u;   // lowest value in monotone-unsigned order
}

// ---------------------------------------------------------------------------
// 1) Wh = x @ W_node^T  via V_WMMA_F32_16X16X4_F32.
//    One wave per 16-row tile; 4 accumulators cover the 64 output columns.
//    A (16x4 f32): lane = M + 16*(K>=2), vgpr = K&1
//    B (4x16 f32): lane = N + 16*(K>=2), vgpr = K&1; B[k][n] = W_node[n*128+k]
//    C (16x16 f32): vgpr j -> M = j (+8 for lanes 16..31), N = lane&15
// ---------------------------------------------------------------------------
__global__ void gat_gemm_wmma(const float* __restrict__ x,
                              const float* __restrict__ Wn,   // [64,128]
                              float* __restrict__ Wh,         // [N,64]
                              int N_, int numTiles) {
    int wave = (int)((blockIdx.x * blockDim.x + threadIdx.x) >> 5);
    int lane = (int)(threadIdx.x & 31);
    if (wave >= numTiles) return;                 // wave-uniform: EXEC stays full

    const int r0 = wave * 16;
    const int m  = lane & 15;
    const int kb = (lane >> 4) << 1;              // 0 or 2

    int rowA = r0 + m; if (rowA >= N_) rowA = N_ - 1;   // clamp (safe garbage)
    const float* xr = x + (size_t)rowA * IN_DIM;
    const float* w0 = Wn + (size_t)( 0 + m) * IN_DIM;
    const float* w1 = Wn + (size_t)(16 + m) * IN_DIM;
    const float* w2 = Wn + (size_t)(32 + m) * IN_DIM;
    const float* w3 = Wn + (size_t)(48 + m) * IN_DIM;

    v8f acc0 = {}, acc1 = {}, acc2 = {}, acc3 = {};

    for (int k0 = 0; k0 < IN_DIM; k0 += 4) {
        const int k = k0 + kb;
        v2f a;  a.x = xr[k];  a.y = xr[k + 1];
        v2f b0; b0.x = w0[k]; b0.y = w0[k + 1];
        v2f b1; b1.x = w1[k]; b1.y = w1[k + 1];
        v2f b2; b2.x = w2[k]; b2.y = w2[k + 1];
        v2f b3; b3.x = w3[k]; b3.y = w3[k + 1];
        acc0 = __builtin_amdgcn_wmma_f32_16x16x4_f32(false, a, false, b0, (short)0, acc0, false, false);
        acc1 = __builtin_amdgcn_wmma_f32_16x16x4_f32(false, a, false, b1, (short)0, acc1, false, false);
        acc2 = __builtin_amdgcn_wmma_f32_16x16x4_f32(false, a, false, b2, (short)0, acc2, false, false);
        acc3 = __builtin_amdgcn_wmma_f32_16x16x4_f32(false, a, false, b3, (short)0, acc3, false, false);
    }

    const int col    = lane & 15;
    const int rowAdd = (lane >> 4) * 8;
    #pragma unroll
    for (int j = 0; j < 8; ++j) {
        int row = r0 + j + rowAdd;
        if (row < N_) {
            float* o = Wh + (size_t)row * OUT_DIM + col;
            o[ 0] = acc0[j];
            o[16] = acc1[j];
            o[32] = acc2[j];
            o[48] = acc3[j];
        }
    }
}

// ---------------------------------------------------------------------------
// 2) per-node attention scalars: a_src[n,h], a_dst[n,h]
// ---------------------------------------------------------------------------
__global__ void gat_node_att(const float* __restrict__ Wh,
                             const float* __restrict__ att_src,  // [4,16]
                             const float* __restrict__ att_dst,  // [4,16]
                             float* __restrict__ a_src,
                             float* __restrict__ a_dst, int NH) {
    int t = blockIdx.x * blockDim.x + threadIdx.x;
    if (t >= NH) return;
    const int h = t & (HEADS - 1);
    const float* w = Wh + (size_t)t * HEAD_DIM;   // t = n*4 + h -> offset n*64 + h*16
    const float* as = att_src + h * HEAD_DIM;
    const float* ad = att_dst + h * HEAD_DIM;
    float ss = 0.f, sd = 0.f;
    #pragma unroll
    for (int i = 0; i < HEAD_DIM; ++i) {
        float v = w[i];
        ss += v * as[i];
        sd += v * ad[i];
    }
    a_src[t] = ss;
    a_dst[t] = sd;
}

// ---------------------------------------------------------------------------
// 3) edge logits + global max: e = leaky_relu(a_src[s] + a_dst[d] + ea@We^T)
// ---------------------------------------------------------------------------
__global__ void gat_edge_logits(const int* __restrict__ ei,
                                const float* __restrict__ ea,
                                const float* __restrict__ We,   // [4,16]
                                const float* __restrict__ a_src,
                                const float* __restrict__ a_dst,
                                float* __restrict__ e_buf,      // [E,4]
                                unsigned* __restrict__ gmax_u, int E_) {
    __shared__ unsigned smax[256];
    int e = blockIdx.x * blockDim.x + threadIdx.x;
    unsigned lmax = 0u;
    if (e < E_) {
        const int s = ei[e];
        const int d = ei[E_ + e];
        const float4* ea4 = (const float4*)(ea + (size_t)e * EDGE_DIM);
        float4 q0 = ea4[0], q1 = ea4[1], q2 = ea4[2], q3 = ea4[3];
        float eav[EDGE_DIM] = { q0.x, q0.y, q0.z, q0.w, q1.x, q1.y, q1.z, q1.w,
                                q2.x, q2.y, q2.z, q2.w, q3.x, q3.y, q3.z, q3.w };
        #pragma unroll
        for (int h = 0; h < HEADS; ++h) {
            float dot = 0.f;
            #pragma unroll
            for (int i = 0; i < EDGE_DIM; ++i) dot += eav[i] * We[h * EDGE_DIM + i];
            float v = a_src[s * HEADS + h] + a_dst[d * HEADS + h] + dot;
            v = (v > 0.f) ? v : 0.2f * v;                 // leaky_relu, slope 0.2
            e_buf[(size_t)e * HEADS + h] = v;
            unsigned ev = enc_f(v);
            lmax = (ev > lmax) ? ev : lmax;
        }
    }
    smax[threadIdx.x] = lmax;
    __syncthreads();
    for (int st = 128; st > 0; st >>= 1) {
        if ((int)threadIdx.x < st) {
            unsigned o = smax[threadIdx.x + st];
            if (o > smax[threadIdx.x]) smax[threadIdx.x] = o;
        }
        __syncthreads();
    }
    if (threadIdx.x == 0) atomicMax(gmax_u, smax[0]);
}

// ---------------------------------------------------------------------------
// 4) ex = exp(e - gmax); den[d,h] += ex  (segment-sum denominators)
// ---------------------------------------------------------------------------
__global__ void gat_edge_exp(const int* __restrict__ ei,
                             float* __restrict__ e_buf,
                             float* __restrict__ den,
                             const unsigned* __restrict__ gmax_u, int E_) {
    int e = blockIdx.x * blockDim.x + threadIdx.x;
    if (e >= E_) return;
    const float gmax = dec_f(*gmax_u);
    const int d = ei[E_ + e];
    #pragma unroll
    for (int h = 0; h < HEADS; ++h) {
        float ex = expf(e_buf[(size_t)e * HEADS + h] - gmax);
        e_buf[(size_t)e * HEADS + h] = ex;
        atomicAdd(den + (size_t)d * HEADS + h, ex);
    }
}

// ---------------------------------------------------------------------------
// 5) message scatter: one wave per edge; lane l owns output elems 2l, 2l+1
//    out[d] += Wh[s] * alpha  (alpha = ex / (den[d] + 1e-9))
// ---------------------------------------------------------------------------
__global__ void gat_edge_scatter(const int* __restrict__ ei,
                                 const float* __restrict__ e_buf,  // ex, [E,4]
                                 const float* __restrict__ den,
                                 const float* __restrict__ Wh,
                                 float* __restrict__ out, int E_) {
    long long gid = (long long)blockIdx.x * blockDim.x + threadIdx.x;
    int e    = (int)(gid >> 5);
    int lane = (int)(gid & 31);
    if (e >= E_) return;
    const int s = ei[e];
    const int d = ei[E_ + e];
    const int h = lane >> 3;                                 // head of elems 2l,2l+1
    const float alpha = e_buf[(size_t)e * HEADS + h]
                      / (den[(size_t)d * HEADS + h] + 1e-9f);
    const float2 w = *(const float2*)(Wh + (size_t)s * OUT_DIM + lane * 2);
    float* o = out + (size_t)d * OUT_DIM + lane * 2;
    atomicAdd(o + 0, w.x * alpha);
    atomicAdd(o + 1, w.y * alpha);
}

// ---------------------------------------------------------------------------
// 6) finalize: out = elu(acc + bias)
// ---------------------------------------------------------------------------
__global__ void gat_finalize(float* __restrict__ out,
                             const float* __restrict__ bias, int total) {
    int t = blockIdx.x * blockDim.x + threadIdx.x;
    if (t >= total) return;
    float v = out[t] + bias[t & (OUT_DIM - 1)];
    out[t] = (v > 0.f) ? v : (expf(v) - 1.0f);
}

// ---------------------------------------------------------------------------
extern "C" void kernel_launch(void* const* d_in, const int* in_sizes, int n_in,
                              void* d_out, int out_size, void* d_ws, size_t ws_size,
                              hipStream_t stream) {
    (void)n_in; (void)out_size; (void)ws_size;

    const float* x       = (const float*)d_in[0];
    const int*   ei      = (const int*)  d_in[1];
    const float* ea      = (const float*)d_in[2];
    const float* Wnode   = (const float*)d_in[3];
    const float* Wedge   = (const float*)d_in[4];
    const float* att_src = (const float*)d_in[5];
    const float* att_dst = (const float*)d_in[6];
    const float* bias    = (const float*)d_in[7];
    float* out = (float*)d_out;

    const int N_ = in_sizes[0] / IN_DIM;
    const int E_ = in_sizes[1] / 2;

    // workspace layout (floats): Wh | a_src | a_dst | den | e_buf | gmax
    float* ws = (float*)d_ws;
    size_t off = 0;
    float* Wh    = ws + off; off += (size_t)N_ * OUT_DIM;
    float* a_src = ws + off; off += (size_t)N_ * HEADS;
    float* a_dst = ws + off; off += (size_t)N_ * HEADS;
    float* den   = ws + off; off += (size_t)N_ * HEADS;
    float* e_buf = ws + off; off += (size_t)E_ * HEADS;
    unsigned* gmax = (unsigned*)(ws + off);

    const int B = 256;

    // 0) zero accumulators
    int initN = N_ * OUT_DIM;
    gat_init<<<(initN + B - 1) / B, B, 0, stream>>>(out, den, gmax, initN, N_ * HEADS);

    // 1) Wh = x @ W_node^T   (WMMA f32 16x16x4)
    int numTiles = (N_ + 15) / 16;
    gat_gemm_wmma<<<(numTiles + 7) / 8, B, 0, stream>>>(x, Wnode, Wh, N_, numTiles);

    // 2) per-node attention scalars
    gat_node_att<<<(N_ * HEADS + B - 1) / B, B, 0, stream>>>(Wh, att_src, att_dst,
                                                             a_src, a_dst, N_ * HEADS);

    // 3) edge logits + global max
    gat_edge_logits<<<(E_ + B - 1) / B, B, 0, stream>>>(ei, ea, Wedge, a_src, a_dst,
                                                        e_buf, gmax, E_);

    // 4) exp + denominators
    gat_edge_exp<<<(E_ + B - 1) / B, B, 0, stream>>>(ei, e_buf, den, gmax, E_);

    // 5) message scatter (wave per edge)
    long long scatterThreads = (long long)E_ * 32;
    gat_edge_scatter<<<(unsigned)((scatterThreads + B - 1) / B), B, 0, stream>>>(
        ei, e_buf, den, Wh, out, E_);

    // 6) elu(acc + bias)
    gat_finalize<<<(N_ * OUT_DIM + B - 1) / B, B, 0, stream>>>(out, bias, N_ * OUT_DIM);
}